// StaticMHCAdapter_84413287236102
// MI455X (gfx1250) — compile-verified
//
#include <hip/hip_runtime.h>
#include <math.h>

typedef __attribute__((ext_vector_type(2))) float v2f;
typedef __attribute__((ext_vector_type(4))) float v4f;
typedef __attribute__((ext_vector_type(8))) float v8f;

#define SINKHORN_ITERS 10
#define INV_TAU 20.0f   // 1 / 0.05

// ---------------------------------------------------------------------------
// Kernel 1: tiny device-side Sinkhorn + softmaxes -> 4x4 mix matrix M in ws.
// M[s][t] = sinkhorn(H_res)[s][t] + softmax(H_pre)[s] * softmax(H_post)[t]
// Trivial scalar work; one thread.
// ---------------------------------------------------------------------------
__device__ __forceinline__ void softmax4(const float* in, float* o) {
    float m = fmaxf(fmaxf(in[0], in[1]), fmaxf(in[2], in[3]));
    float s = 0.0f;
    #pragma unroll
    for (int i = 0; i < 4; ++i) { o[i] = __expf(in[i] - m); s += o[i]; }
    float inv = 1.0f / s;
    #pragma unroll
    for (int i = 0; i < 4; ++i) o[i] *= inv;
}

__global__ void mhc_mix_setup(const float* __restrict__ res_logits,
                              const float* __restrict__ pre_logits,
                              const float* __restrict__ post_logits,
                              float* __restrict__ M) {
    if (threadIdx.x != 0 || blockIdx.x != 0) return;
    float z[4][4], u[4] = {0,0,0,0}, v[4] = {0,0,0,0};
    #pragma unroll
    for (int s = 0; s < 4; ++s)
        #pragma unroll
        for (int t = 0; t < 4; ++t)
            z[s][t] = res_logits[s*4 + t] * INV_TAU;

    for (int it = 0; it < SINKHORN_ITERS; ++it) {
        // u = -logsumexp over t of (z + v[t])
        #pragma unroll
        for (int s = 0; s < 4; ++s) {
            float m = -INFINITY;
            #pragma unroll
            for (int t = 0; t < 4; ++t) m = fmaxf(m, z[s][t] + v[t]);
            float acc = 0.0f;
            #pragma unroll
            for (int t = 0; t < 4; ++t) acc += __expf(z[s][t] + v[t] - m);
            u[s] = -(m + __logf(acc));
        }
        // v = -logsumexp over s of (z + u[s])   (uses the new u)
        #pragma unroll
        for (int t = 0; t < 4; ++t) {
            float m = -INFINITY;
            #pragma unroll
            for (int s = 0; s < 4; ++s) m = fmaxf(m, z[s][t] + u[s]);
            float acc = 0.0f;
            #pragma unroll
            for (int s = 0; s < 4; ++s) acc += __expf(z[s][t] + u[s] - m);
            v[t] = -(m + __logf(acc));
        }
    }
    float hpre[4], hpost[4];
    softmax4(pre_logits, hpre);
    softmax4(post_logits, hpost);
    #pragma unroll
    for (int s = 0; s < 4; ++s)
        #pragma unroll
        for (int t = 0; t < 4; ++t)
            M[s*4 + t] = __expf(z[s][t] + u[s] + v[t]) + hpre[s] * hpost[t];
}

// ---------------------------------------------------------------------------
// Kernel 2: bandwidth-streaming stream-mix via V_WMMA_F32_16X16X4_F32.
//
// One wave per "tile" = 64 consecutive elements of one batch, all 4 streams.
// WMMA j (j=0..3) handles elements {e0 + 4m + j : m=0..15}:
//   A_j[m,k] = x[stream k, e0+4m+j]   (lanes 0-15: k=0,1 in v0,v1;
//                                      lanes 16-31: k=2,3 — ISA A 16x4 layout)
//   B[k,n]   = M[k][n] for n<4 else 0 (lanes 0-15: k=0,1; lanes 16-31: k=2,3)
//   D_j[m,n] = out[stream n, e0+4m+j]
// A loads: 2 full-wave nontemporal B128 loads (1 KB, fully used).
// D stores: lane n (n<4) / lane 16+n hold 32 consecutive floats of stream n
//           -> 8 nontemporal B128 stores, 16B-contiguous per lane.
// ---------------------------------------------------------------------------
__global__ void mhc_mix_apply(const float* __restrict__ x,
                              float* __restrict__ out,
                              const float* __restrict__ M,
                              long long Ne,          // elements per (b,s) plane
                              int chunks,            // Ne / 64
                              int total_tiles) {     // 4 * chunks
    const int lane  = threadIdx.x & 31;
    const int lm    = lane & 15;
    const bool hi   = lane >= 16;
    const int waves_per_blk = blockDim.x >> 5;
    int gwave  = blockIdx.x * waves_per_blk + (threadIdx.x >> 5);
    int nwaves = gridDim.x * waves_per_blk;

    // Constant B fragment (4x16, M in columns 0..3).
    v2f bm = {0.0f, 0.0f};
    if (lm < 4) {
        bm.x = M[(hi ? 2 : 0) * 4 + lm];   // K = 0 or 2
        bm.y = M[(hi ? 3 : 1) * 4 + lm];   // K = 1 or 3
    }
    const long long soff = hi ? 2 * Ne : 0;   // stream offset for this half-wave

    for (int t = gwave; t < total_tiles; t += nwaves) {
        const int b = t / chunks;
        const int c = t - b * chunks;
        const long long e0 = (long long)c * 64;
        const float* bp = x + (long long)b * 4 * Ne + e0 + soff + 4 * lm;

        // Prefetch next tile's A data (speculative; dropped if stale).
        int tn = t + nwaves;
        if (tn < total_tiles) {
            int bn = tn / chunks;
            int cn = tn - bn * chunks;
            const float* pp = x + (long long)bn * 4 * Ne + (long long)cn * 64 + soff + 4 * lm;
            __builtin_prefetch(pp, 0, 0);
            __builtin_prefetch(pp + Ne, 0, 0);
        }

        // Streams {0,2} and {1,3}: two full-wave 128-bit nontemporal loads.
        v4f a0 = __builtin_nontemporal_load((const v4f*)bp);        // K = 0 / 2
        v4f a1 = __builtin_nontemporal_load((const v4f*)(bp + Ne)); // K = 1 / 3

        v8f cz = {};
        v2f aa;
        aa.x = a0.x; aa.y = a1.x;
        v8f d0 = __builtin_amdgcn_wmma_f32_16x16x4_f32(false, aa, false, bm, (short)0, cz, false, false);
        aa.x = a0.y; aa.y = a1.y;
        v8f d1 = __builtin_amdgcn_wmma_f32_16x16x4_f32(false, aa, false, bm, (short)0, cz, false, false);
        aa.x = a0.z; aa.y = a1.z;
        v8f d2 = __builtin_amdgcn_wmma_f32_16x16x4_f32(false, aa, false, bm, (short)0, cz, false, false);
        aa.x = a0.w; aa.y = a1.w;
        v8f d3 = __builtin_amdgcn_wmma_f32_16x16x4_f32(false, aa, false, bm, (short)0, cz, false, false);

        // Lane n<4 (resp. 16+n) holds out[stream n, e0 + (hi?32:0) + 4r + j] = d_j[r].
        if (lm < 4) {
            float* op = out + (long long)b * 4 * Ne + (long long)lm * Ne + e0 + (hi ? 32 : 0);
            #pragma unroll
            for (int r = 0; r < 8; ++r) {
                v4f w;
                w.x = d0[r]; w.y = d1[r]; w.z = d2[r]; w.w = d3[r];
                __builtin_nontemporal_store(w, (v4f*)(op + 4 * r));
            }
        }
    }
}

// ---------------------------------------------------------------------------
extern "C" void kernel_launch(void* const* d_in, const int* in_sizes, int n_in,
                              void* d_out, int out_size, void* d_ws, size_t ws_size,
                              hipStream_t stream) {
    const float* x     = (const float*)d_in[0];   // [16, T, D] f32
    const float* hres  = (const float*)d_in[1];   // [4,4]
    const float* hpre  = (const float*)d_in[2];   // [4]
    const float* hpost = (const float*)d_in[3];   // [4]
    float* out = (float*)d_out;
    float* M   = (float*)d_ws;                    // 16 floats of scratch

    mhc_mix_setup<<<1, 32, 0, stream>>>(hres, hpre, hpost, M);

    const long long total = (long long)in_sizes[0];   // 16 * T * D
    const long long Ne    = total / 16;               // per-(b,s) plane elements
    const int chunks      = (int)(Ne / 64);           // 64-element tiles per plane
    const int total_tiles = 4 * chunks;               // over B=4 batches

    int blocks = 2048;                                // 16384 waves, grid-stride
    int max_blocks = (total_tiles + 7) / 8;           // 8 waves per block
    if (blocks > max_blocks) blocks = max_blocks;
    if (blocks < 1) blocks = 1;

    mhc_mix_apply<<<blocks, 256, 0, stream>>>(x, out, M, Ne, chunks, total_tiles);
}